// GaussianRenderer_68453188763650
// MI455X (gfx1250) — compile-verified
//
#include <hip/hip_runtime.h>

#define HH 256
#define WW 256
#define NG 512
#define EPSV 1e-4f

typedef float v2f __attribute__((ext_vector_type(2)));
typedef float v8f __attribute__((ext_vector_type(8)));

// -0.5 * log2(e): folds exp(-0.5*mahal) into exp2(dot(q, basis))
#define SFOLD (-0.72134752044448170f)

// ---------------------------------------------------------------------------
// Kernel 1: per-gaussian projection + covariance + depth rank-sort + scatter
// of quadratic-form coefficients into depth-sorted SoA arrays in workspace.
// ---------------------------------------------------------------------------
__global__ __launch_bounds__(NG) void gs_prep_kernel(
    const float* __restrict__ means3D, const float* __restrict__ covs3d,
    const float* __restrict__ colors,  const float* __restrict__ opac,
    const float* __restrict__ Km, const float* __restrict__ Rm,
    const float* __restrict__ tv,
    float2* __restrict__ q01, float2* __restrict__ q23,
    float2* __restrict__ q45, float4* __restrict__ colw)
{
    __shared__ float sdepth[NG];
    const int g = threadIdx.x;

    // Camera constants (uniform -> scalar loads)
    float Rr[3][3];
#pragma unroll
    for (int i = 0; i < 3; ++i)
#pragma unroll
        for (int j = 0; j < 3; ++j) Rr[i][j] = Rm[3 * i + j];
    const float fx = Km[0], cx = Km[2], fy = Km[4], cy = Km[5];
    const float t0 = tv[0], t1 = tv[1], t2 = tv[2];

    const float m0 = means3D[3 * g + 0];
    const float m1 = means3D[3 * g + 1];
    const float m2 = means3D[3 * g + 2];

    // cam = R * m + t
    const float cam0 = Rr[0][0] * m0 + Rr[0][1] * m1 + Rr[0][2] * m2 + t0;
    const float cam1 = Rr[1][0] * m0 + Rr[1][1] * m1 + Rr[1][2] * m2 + t1;
    const float cam2 = Rr[2][0] * m0 + Rr[2][1] * m1 + Rr[2][2] * m2 + t2;

    const float z = cam2;
    const float depth = fmaxf(z, 1.0f);
    sdepth[g] = depth;

    // mean2D
    const float inv_s2 = 1.0f / cam2;
    const float mx = (fx * cam0 + cx * cam2) * inv_s2;
    const float my = (fy * cam1 + cy * cam2) * inv_s2;

    // Jacobian rows u=(j00,0,j02), v=(0,j11,j12)
    const float iz = 1.0f / z, iz2 = iz * iz;
    const float j00 = fx * iz, j02 = -fx * cam0 * iz2;
    const float j11 = fy * iz, j12 = -fy * cam1 * iz2;

    // V = R * Sigma * R^T
    const float* S = covs3d + 9 * g;
    float RS[3][3], V[3][3];
#pragma unroll
    for (int i = 0; i < 3; ++i)
#pragma unroll
        for (int j = 0; j < 3; ++j)
            RS[i][j] = Rr[i][0] * S[0 * 3 + j] + Rr[i][1] * S[1 * 3 + j] + Rr[i][2] * S[2 * 3 + j];
#pragma unroll
    for (int i = 0; i < 3; ++i)
#pragma unroll
        for (int j = 0; j < 3; ++j)
            V[i][j] = RS[i][0] * Rr[j][0] + RS[i][1] * Rr[j][1] + RS[i][2] * Rr[j][2];

    // cov2D = J V J^T (2x2)
    const float Vu0 = V[0][0] * j00 + V[0][2] * j02;
    const float Vu2 = V[2][0] * j00 + V[2][2] * j02;
    const float Vv0 = V[0][1] * j11 + V[0][2] * j12;
    const float Vv1 = V[1][1] * j11 + V[1][2] * j12;
    const float Vv2 = V[2][1] * j11 + V[2][2] * j12;
    const float c00 = j00 * Vu0 + j02 * Vu2 + EPSV;
    const float c01 = j00 * Vv0 + j02 * Vv2;
    const float c11 = j11 * Vv1 + j12 * Vv2 + EPSV;

    const float det  = c00 * c11 - c01 * c01;
    const float idet = 1.0f / det;
    const float i00 = c11 * idet, i01 = -c01 * idet, i11 = c00 * idet;

    const float norm  = 0.15915494309189535f * rsqrtf(det); // 1/(2*pi*sqrt(det))
    const bool  valid = (depth > 1.0f) && (depth < 50.0f);
    const float w     = valid ? (opac[g] * norm) : 0.0f;

    // Quadratic-form coefficients (exp2-folded):
    // E(x,y) = q0*x^2 + q1*x*y + q2*y^2 + q3*x + q4*y + q5  =  -0.5*log2e*mahal
    const float q0 = SFOLD * i00;
    const float q1 = SFOLD * 2.0f * i01;
    const float q2 = SFOLD * i11;
    const float q3 = SFOLD * (-2.0f) * (i00 * mx + i01 * my);
    const float q4 = SFOLD * (-2.0f) * (i01 * mx + i11 * my);
    const float q5 = SFOLD * (i00 * mx * mx + 2.0f * i01 * mx * my + i11 * my * my);

    __syncthreads();

    // Stable rank by depth (matches jnp.argsort)
    int rank = 0;
    const float dg = depth;
    for (int j = 0; j < NG; ++j) {
        const float dj = sdepth[j];
        rank += (dj < dg) || ((dj == dg) && (j < g));
    }

    q01[rank]  = make_float2(q0, q1);
    q23[rank]  = make_float2(q2, q3);
    q45[rank]  = make_float2(q4, q5);
    colw[rank] = make_float4(colors[3 * g], colors[3 * g + 1], colors[3 * g + 2], w);
}

// ---------------------------------------------------------------------------
// Kernel 2: WMMA exponent tiles + per-lane front-to-back compositing.
// Each wave owns a 32x1 pixel strip; block = 8 waves = 32x8 pixel tile.
// D tile: M = 16 gaussians, N = 16 pixels; two N-tiles per wave.
// ---------------------------------------------------------------------------
__global__ __launch_bounds__(256) void gs_render_kernel(
    const float2* __restrict__ q01, const float2* __restrict__ q23,
    const float2* __restrict__ q45, const float4* __restrict__ colw,
    float* __restrict__ out)
{
    // stride 34 rows -> conflict-free LDS write (WMMA layout) & read (per-lane)
    __shared__ float ldsE[8][16][34];

    const int lane = threadIdx.x & 31;
    const int wave = threadIdx.x >> 5;
    const int half = lane & 15;
    const bool hi  = lane >= 16;

    const int py  = blockIdx.y * 8 + wave;
    const int px0 = blockIdx.x * 32 + half;
    const float X0 = (float)px0, X1 = (float)(px0 + 16), Y = (float)py;

    // B operands (4x16 f32): lanes 0-15 hold K=0..1, lanes 16-31 hold K=2..3.
    // Feature order: WMMA#1 K = [x^2, x*y, y^2, x]; WMMA#2 K = [y, 1, 0, 0].
    v2f b1t0, b1t1, b2;
    b1t0.x = hi ? (Y * Y) : (X0 * X0);
    b1t0.y = hi ? X0      : (X0 * Y);
    b1t1.x = hi ? (Y * Y) : (X1 * X1);
    b1t1.y = hi ? X1      : (X1 * Y);
    b2.x   = hi ? 0.0f    : Y;
    b2.y   = hi ? 0.0f    : 1.0f;

    float T = 1.0f, cr = 0.0f, cg = 0.0f, cb = 0.0f;

    for (int base = 0; base < NG; base += 16) {
        // A operands (16x4 f32): row M = gaussian, lanes 0-15 K=0..1, 16-31 K=2..3
        const int gl = base + half;
        const float2 p01 = q01[gl];
        const float2 p23 = q23[gl];
        const float2 p45 = q45[gl];
        v2f a1, a2;
        a1.x = hi ? p23.x : p01.x;
        a1.y = hi ? p23.y : p01.y;
        a2.x = hi ? 0.0f  : p45.x;
        a2.y = hi ? 0.0f  : p45.y;

        // E = A(16g x 8) * B(8 x 16px), chained over two K=4 WMMAs
        v8f d0 = {};
        d0 = __builtin_amdgcn_wmma_f32_16x16x4_f32(false, a1, false, b1t0,
                                                   (short)0, d0, false, false);
        d0 = __builtin_amdgcn_wmma_f32_16x16x4_f32(false, a2, false, b2,
                                                   (short)0, d0, false, false);
        v8f d1 = {};
        d1 = __builtin_amdgcn_wmma_f32_16x16x4_f32(false, a1, false, b1t1,
                                                   (short)0, d1, false, false);
        d1 = __builtin_amdgcn_wmma_f32_16x16x4_f32(false, a2, false, b2,
                                                   (short)0, d1, false, false);

        // Spill D tiles: row = gaussian (M = vgpr + 8*hi), col = pixel
        const int gr = hi ? 8 : 0;
#pragma unroll
        for (int i = 0; i < 8; ++i) {
            ldsE[wave][gr + i][half]      = d0[i];
            ldsE[wave][gr + i][half + 16] = d1[i];
        }

        // Composite: each lane owns pixel x = blockIdx.x*32 + lane, scans
        // the 16 gaussians in sorted (depth) order. colw[] index is wave-
        // uniform -> scalar loads.
#pragma unroll
        for (int i = 0; i < 16; ++i) {
            const float  e   = ldsE[wave][i][lane];
            const float4 cwv = colw[base + i];
            const float  a   = cwv.w * __builtin_amdgcn_exp2f(e);
            const float  aT  = a * T;
            cr = fmaf(aT, cwv.x, cr);
            cg = fmaf(aT, cwv.y, cg);
            cb = fmaf(aT, cwv.z, cb);
            T  = T * (1.0f - a);
        }
    }

    const int x = blockIdx.x * 32 + lane;
    const int o = (py * WW + x) * 3;
    out[o + 0] = cr;
    out[o + 1] = cg;
    out[o + 2] = cb;
}

// ---------------------------------------------------------------------------
extern "C" void kernel_launch(void* const* d_in, const int* in_sizes, int n_in,
                              void* d_out, int out_size, void* d_ws, size_t ws_size,
                              hipStream_t stream) {
    const float* means3D = (const float*)d_in[0];
    const float* covs3d  = (const float*)d_in[1];
    const float* colors  = (const float*)d_in[2];
    const float* opac    = (const float*)d_in[3];
    const float* Km      = (const float*)d_in[4];
    const float* Rm      = (const float*)d_in[5];
    const float* tv      = (const float*)d_in[6];

    char* ws = (char*)d_ws;
    float2* q01  = (float2*)(ws + 0);
    float2* q23  = (float2*)(ws + 4096);
    float2* q45  = (float2*)(ws + 8192);
    float4* colw = (float4*)(ws + 12288);   // 20 KB total

    gs_prep_kernel<<<1, NG, 0, stream>>>(means3D, covs3d, colors, opac,
                                         Km, Rm, tv, q01, q23, q45, colw);

    dim3 grid(WW / 32, HH / 8);
    gs_render_kernel<<<grid, 256, 0, stream>>>(q01, q23, q45, colw, (float*)d_out);
}